// Net_13743895347756
// MI455X (gfx1250) — compile-verified
//
#include <hip/hip_runtime.h>

typedef float v2f __attribute__((ext_vector_type(2)));
typedef float v8f __attribute__((ext_vector_type(8)));

#define WMMA_F32(a, b, c) \
  __builtin_amdgcn_wmma_f32_16x16x4_f32(false, (a), false, (b), (short)0, (c), false, false)

static constexpr int NN = 2560;   // nodes
static constexpr int EE = 4096;   // edges

// ---------------------------------------------------------------------------
__global__ void zero_kernel(float* __restrict__ p, int n) {
  int i = blockIdx.x * blockDim.x + threadIdx.x;
  if (i < n) p[i] = 0.f;
}

// ---------------------------------------------------------------------------
// Edge MLP: h2e = relu(relu(ea@w1+b1)@w2+b2), ea:[E,3] -> h2e:[E,64]
// 128 threads (4 waves) per block, 64 edges per block (16 per wave).
__global__ void edge_mlp_kernel(const float* __restrict__ ea,
                                const float* __restrict__ w1, const float* __restrict__ b1,
                                const float* __restrict__ w2, const float* __restrict__ b2,
                                float* __restrict__ h2e) {
  __shared__ float sea[64 * 3];
  __shared__ float sw1[3 * 64];
  __shared__ float sb1[64];
  __shared__ float sw2[64 * 64];
  __shared__ float sb2[64];
  __shared__ float sh1[64 * 64];
  const int tid = threadIdx.x, nthr = blockDim.x;
  const int ebase = blockIdx.x * 64;
  for (int t = tid; t < 64 * 3; t += nthr) sea[t] = ea[ebase * 3 + t];
  for (int t = tid; t < 3 * 64; t += nthr) sw1[t] = w1[t];
  for (int t = tid; t < 64; t += nthr) { sb1[t] = b1[t]; sb2[t] = b2[t]; }
  for (int t = tid; t < 64 * 64; t += nthr) sw2[t] = w2[t];
  __syncthreads();
  // layer 1 (K=3): scalar
  for (int t = tid; t < 64 * 64; t += nthr) {
    int e = t >> 6, j = t & 63;
    float s = sb1[j] + sea[e * 3 + 0] * sw1[j] + sea[e * 3 + 1] * sw1[64 + j] +
              sea[e * 3 + 2] * sw1[128 + j];
    sh1[e * 64 + j] = s > 0.f ? s : 0.f;
  }
  __syncthreads();
  // layer 2: [16,64]@[64,64] per wave via WMMA f32 16x16x4
  const int wave = tid >> 5, lane = tid & 31;
  const int col = lane & 15, hi = lane >> 4;
  const int erow = wave * 16;
  for (int nt = 0; nt < 4; ++nt) {
    v8f acc = {};
    for (int k0 = 0; k0 < 64; k0 += 4) {
      v2f a, b;
      a[0] = sh1[(erow + col) * 64 + k0 + 2 * hi];
      a[1] = sh1[(erow + col) * 64 + k0 + 1 + 2 * hi];
      b[0] = sw2[(k0 + 2 * hi) * 64 + nt * 16 + col];
      b[1] = sw2[(k0 + 1 + 2 * hi) * 64 + nt * 16 + col];
      acc = WMMA_F32(a, b, acc);
    }
    float bb = sb2[nt * 16 + col];
#pragma unroll
    for (int r = 0; r < 8; ++r) {
      int row = r + 8 * hi;
      float v = acc[r] + bb;
      h2e[(ebase + erow + row) * 64 + nt * 16 + col] = v > 0.f ? v : 0.f;
    }
  }
}

// ---------------------------------------------------------------------------
// Fused NNConv message pass: W[e] = h2e[e]@w3 + b3 is generated in the WMMA
// accumulator (never materialized: 268MB for layer 3 would overflow L2);
// msg[e,o] = sum_i h[src,i]*W[e,i,o]; atomic scatter-add into agg[dst].
// ET edge tiles (ET*16 edges) per block so each w3 B-fragment global load
// feeds ET back-to-back WMMAs (B-operand reuse, 1/ET the L2 traffic).
template <int ET>
__global__ void nnconv_msg_kernel(const float* __restrict__ h2e,   // [E,64]
                                  const float* __restrict__ w3,    // [64, din*dout]
                                  const float* __restrict__ b3,    // [din*dout]
                                  const float* __restrict__ h,     // [N, din]
                                  const int* __restrict__ src, const int* __restrict__ dst,
                                  float* __restrict__ agg,         // [N, dout]
                                  int din, int dout) {
  extern __shared__ float smem[];
  float* s_h2e = smem;                    // ET*16*64
  float* s_hsrc = smem + ET * 16 * 64;    // ET*16*din
  __shared__ int s_dst[ET * 16];
  const int tid = threadIdx.x, nthr = blockDim.x;
  const int ebase = blockIdx.x * (16 * ET);
  for (int t = tid; t < ET * 16 * 64; t += nthr) s_h2e[t] = h2e[ebase * 64 + t];
  for (int t = tid; t < ET * 16; t += nthr) s_dst[t] = dst[ebase + t];
  for (int t = tid; t < ET * 16 * din; t += nthr) {
    int e = t / din, i = t - e * din;
    s_hsrc[t] = h[src[ebase + e] * din + i];
  }
  __syncthreads();
  const int wave = tid >> 5, lane = tid & 31;
  const int col = lane & 15, hi = lane >> 4;
  const int nbase = wave * 16;
  const long ldw = (long)din * dout;
  v8f msg[ET];
#pragma unroll
  for (int t = 0; t < ET; ++t) msg[t] = {};
  for (int i = 0; i < din; ++i) {
    v8f wacc[ET];
#pragma unroll
    for (int t = 0; t < ET; ++t) wacc[t] = {};
    const float* w3c = w3 + (long)i * dout + nbase + col;
    for (int k0 = 0; k0 < 64; k0 += 4) {
      v2f b;
      b[0] = w3c[(k0 + 2 * hi) * ldw];
      b[1] = w3c[(k0 + 1 + 2 * hi) * ldw];
#pragma unroll
      for (int t = 0; t < ET; ++t) {
        v2f a;
        a[0] = s_h2e[(t * 16 + col) * 64 + k0 + 2 * hi];
        a[1] = s_h2e[(t * 16 + col) * 64 + k0 + 1 + 2 * hi];
        wacc[t] = WMMA_F32(a, b, wacc[t]);
      }
    }
    float bv = b3[(long)i * dout + nbase + col];
#pragma unroll
    for (int t = 0; t < ET; ++t) {
#pragma unroll
      for (int r = 0; r < 8; ++r) {
        int row = r + 8 * hi;
        msg[t][r] += s_hsrc[(t * 16 + row) * din + i] * (wacc[t][r] + bv);
      }
    }
  }
#pragma unroll
  for (int t = 0; t < ET; ++t) {
#pragma unroll
    for (int r = 0; r < 8; ++r) {
      int row = r + 8 * hi;
      atomicAdd(&agg[s_dst[t * 16 + row] * dout + nbase + col], msg[t][r]);
    }
  }
}

// ---------------------------------------------------------------------------
// Generic GEMM: out[.., coloff+n] = act(A[M,K]@Wt[K,Nn] + bias + addend[M,Nn]).
// One wave per 16x16 tile, 4 waves/block.
__global__ void gemm_kernel(const float* __restrict__ A, int lda,
                            const float* __restrict__ Wt, int ldw,
                            const float* __restrict__ bias,
                            const float* __restrict__ addend,
                            float* __restrict__ out, int ldo, int coloff,
                            int M, int K, int Nn, int do_relu) {
  const int tilesN = Nn >> 4;
  const int tile = blockIdx.x * (blockDim.x >> 5) + (threadIdx.x >> 5);
  if (tile >= (M >> 4) * tilesN) return;  // wave-uniform exit (EXEC stays full for WMMA)
  const int tm = tile / tilesN, tn = tile - tm * tilesN;
  const int mbase = tm * 16, nbase = tn * 16;
  const int lane = threadIdx.x & 31;
  const int col = lane & 15, hi = lane >> 4;
  v8f acc = {};
  const float* Ar = A + (mbase + col) * lda;
  for (int k0 = 0; k0 < K; k0 += 4) {
    v2f a, b;
    a[0] = Ar[k0 + 2 * hi];
    a[1] = Ar[k0 + 1 + 2 * hi];
    b[0] = Wt[(k0 + 2 * hi) * ldw + nbase + col];
    b[1] = Wt[(k0 + 1 + 2 * hi) * ldw + nbase + col];
    acc = WMMA_F32(a, b, acc);
  }
  float bv = bias ? bias[nbase + col] : 0.f;
#pragma unroll
  for (int r = 0; r < 8; ++r) {
    int row = mbase + r + 8 * hi;
    float v = acc[r] + bv;
    if (addend) v += addend[row * Nn + nbase + col];
    if (do_relu) v = v > 0.f ? v : 0.f;
    out[row * ldo + coloff + nbase + col] = v;
  }
}

// ---------------------------------------------------------------------------
// Head: feat = concat(x3[10g], x3[10g+1]) @ wout + bout  -> [256,1]
__global__ void head_kernel(const float* __restrict__ x3,    // [N,64]
                            const float* __restrict__ wout,  // [128]
                            const float* __restrict__ bout,  // [1]
                            float* __restrict__ out) {
  int g = blockIdx.x * blockDim.x + threadIdx.x;
  if (g >= 256) return;
  const float* a = x3 + (g * 10) * 64;
  const float* c = x3 + (g * 10 + 1) * 64;
  float s = bout[0];
  for (int j = 0; j < 64; ++j) s += a[j] * wout[j] + c[j] * wout[64 + j];
  out[g] = s;
}

// ---------------------------------------------------------------------------
extern "C" void kernel_launch(void* const* d_in, const int* in_sizes, int n_in,
                              void* d_out, int out_size, void* d_ws, size_t ws_size,
                              hipStream_t stream) {
  (void)in_sizes; (void)n_in; (void)out_size; (void)ws_size;
  const float* x = (const float*)d_in[0];
  const int* eidx = (const int*)d_in[1];
  const int* src = eidx;
  const int* dstp = eidx + EE;
  const float* ea[2] = {(const float*)d_in[2], (const float*)d_in[3]};
  auto F = [&](int i) { return (const float*)d_in[i]; };
  // params flattened (JAX sorted-key pytree): per conv: bias,b1,b2,b3,w1,w2,w3,root
  const int dims[4][2] = {{4, 64}, {64, 128}, {128, 128}, {128, 64}};

  float* W = (float*)d_ws;
  float* hA = W;
  float* hB = W + NN * 128;
  float* agg = W + 2 * NN * 128;
  float* xcat = W + 3 * NN * 128;
  float* t1 = W + 4 * NN * 128;
  float* t2 = W + 5 * NN * 128;
  float* t3 = t2 + NN * 64;
  float* h2e = t3 + NN * 64;

  constexpr int ET = 4;  // edge tiles per message-pass block

  for (int br = 0; br < 2; ++br) {
    const int base = 5 + br * 32;
    const float* hcur = x;
    float* hbufs[2] = {hA, hB};
    int cur = 0;
    for (int l = 0; l < 4; ++l) {
      const int din = dims[l][0], dout = dims[l][1];
      const float* bias = F(base + 8 * l + 0);
      const float* b1 = F(base + 8 * l + 1);
      const float* b2 = F(base + 8 * l + 2);
      const float* b3 = F(base + 8 * l + 3);
      const float* w1 = F(base + 8 * l + 4);
      const float* w2 = F(base + 8 * l + 5);
      const float* w3 = F(base + 8 * l + 6);
      const float* root = F(base + 8 * l + 7);

      edge_mlp_kernel<<<EE / 64, 128, 0, stream>>>(ea[br], w1, b1, w2, b2, h2e);

      const int aggn = NN * dout;
      zero_kernel<<<(aggn + 255) / 256, 256, 0, stream>>>(agg, aggn);

      const int nw = dout / 16;
      const size_t lds = (size_t)(ET * 16 * 64 + ET * 16 * din) * sizeof(float);
      nnconv_msg_kernel<ET><<<EE / (16 * ET), nw * 32, lds, stream>>>(
          h2e, w3, b3, hcur, src, dstp, agg, din, dout);

      float* outp; int ldo, coloff;
      if (l == 3) { outp = xcat; ldo = 128; coloff = br * 64; }
      else        { outp = hbufs[cur ^ 1]; ldo = dout; coloff = 0; }
      const int tiles = (NN / 16) * (dout / 16);
      gemm_kernel<<<(tiles + 3) / 4, 128, 0, stream>>>(hcur, din, root, dout, bias, agg,
                                                       outp, ldo, coloff, NN, din, dout, 1);
      if (l < 3) { hcur = hbufs[cur ^ 1]; cur ^= 1; }
    }
  }
  // head: lin1 (128->128), lin2 (128->64), lin3 (64->64), out
  {
    int tiles = (NN / 16) * (128 / 16);
    gemm_kernel<<<(tiles + 3) / 4, 128, 0, stream>>>(xcat, 128, F(70), 128, F(69), nullptr,
                                                     t1, 128, 0, NN, 128, 128, 0);
    tiles = (NN / 16) * (64 / 16);
    gemm_kernel<<<(tiles + 3) / 4, 128, 0, stream>>>(t1, 128, F(72), 64, F(71), nullptr,
                                                     t2, 64, 0, NN, 128, 64, 0);
    gemm_kernel<<<(tiles + 3) / 4, 128, 0, stream>>>(t2, 64, F(74), 64, F(73), nullptr,
                                                     t3, 64, 0, NN, 64, 64, 0);
    head_kernel<<<1, 256, 0, stream>>>(t3, F(76), F(75), (float*)d_out);
  }
}